// HashTables_71184787964443
// MI455X (gfx1250) — compile-verified
//
#include <hip/hip_runtime.h>
#include <stdint.h>

// Embedding hash-gather for MI455X (gfx1250), TDM edition.
//   tokens --async--> LDS --(u32 hash, low 19 bits exact)--> bucket ids
//   bucket ids --v_readlane--> TDM gather descriptors (8 rows x 256B each)
//   tensor_load_to_lds (gather mode) : tables --DMA--> LDS
//   global_store_async_from_lds_b128 : LDS --async NT--> out
// Data never transits VGPRs for the 64MB gather + 64MB store streams.

typedef float    f4   __attribute__((ext_vector_type(4)));
typedef unsigned u32;
typedef u32      u32x4 __attribute__((ext_vector_type(4)));
typedef u32      u32x8 __attribute__((ext_vector_type(8)));

constexpr int      NUM_TABLES = 8;
constexpr unsigned BUCKETS    = 524288u;        // 2^19
constexpr unsigned BMASK      = BUCKETS - 1u;
constexpr int      EMBED      = 64;             // floats per row (256 B)
constexpr int      Bdim       = 8;
constexpr int      Tdim       = 4096;
constexpr int      MAXW       = 128;            // largest window
constexpr int      TILE_T     = 32;             // t-positions per block
constexpr int      BLOCK      = 256;            // 8 waves
constexpr int      NROWS      = TILE_T * NUM_TABLES;        // 256 rows / block
constexpr int      NCHUNK     = NROWS * (EMBED / 4);        // 4096 x 16B

__global__ __launch_bounds__(BLOCK)
void HashTables_tdm_kernel(const int* __restrict__ tokens,
                           const float* __restrict__ tables,
                           float* __restrict__ out) {
    __shared__ int      s_tok[TILE_T + MAXW];                 // 160 ints
    __shared__ unsigned s_idx[NROWS];                         // 256 ids
    __shared__ __align__(256) unsigned char s_buf[NROWS * 256]; // 64 KB tile

    const int tid       = threadIdx.x;
    const int tilesPerB = Tdim / TILE_T;                      // 128
    const int b         = blockIdx.x / tilesPerB;
    const int t0        = (blockIdx.x % tilesPerB) * TILE_T;
    const size_t tokBase = (size_t)b * Tdim;

    // ---- Stage tokens[t0-128 .. t0+TILE_T) into LDS (async global->LDS).
    if (tid < TILE_T + MAXW) {
        int g = t0 - MAXW + tid;
        unsigned ldsOff = (unsigned)(uintptr_t)(&s_tok[tid]); // LDS byte offset
        if (g >= 0) {
            const int* gp = tokens + tokBase + (size_t)g;
            asm volatile("global_load_async_to_lds_b32 %0, %1, off"
                         :: "v"(ldsOff), "v"(gp) : "memory");
        } else {
            s_tok[tid] = 0;     // out-of-range shift contributes token 0
        }
    }
    asm volatile("s_wait_asynccnt 0x0" ::: "memory");
    __syncthreads();

    // ---- Rolling XOR hash; snapshots at windows 1,2,4,...,128.
    // BUCKETS = 2^19 and XOR is bitwise => uint32 math is exact mod 2^19.
    if (tid < TILE_T) {
        const unsigned P[8] = {2654435761u, 2246822519u, 3266489917u,
                               2028178513u, 1220703125u, 1610612741u,
                               805306457u,  402653189u};
        unsigned h = 0u;
        unsigned* dst = &s_idx[tid * NUM_TABLES];
        int k = 0;
#pragma unroll
        for (int j = 0; j < MAXW; ++j) {
            unsigned tok = (unsigned)s_tok[MAXW + tid - 1 - j];
            h ^= tok * P[j & 7];
            if (((j + 1) & j) == 0) dst[k++] = h & BMASK;
        }
    }
    __syncthreads();

    // ---- TDM gather: 32 descriptors/block, 8 rows x 256B each.
    // Descriptor d = tgrp*8 + i gathers rows t = t0+tgrp*8+r (r=0..7) of
    // table i into s_buf + d*2048 (row r at +r*256).
    const unsigned sbufBase = (unsigned)(uintptr_t)(&s_buf[0]);
    const int wv    = __builtin_amdgcn_readfirstlane(tid >> 5); // wave id 0..7
    const int lane  = tid & 31;
    const int tgrp  = wv >> 1;            // 0..3 : group of 8 t-positions
    const int ibase = (wv & 1) * 4;       // tables 0..3 or 4..7
    // Lane l holds index for (r = l>>2, table = ibase + (l&3)).
    unsigned ia = s_idx[(((tgrp << 3) | (lane >> 2)) << 3) | (ibase + (lane & 3))];

    // D# group 1 (shared by all descriptors): data_size=4B, dim0=64,
    // dim1=2^19 (hi16=8), tile_dim0=64, tile_dim1=8 idx, stride0=64.
    const u32x8 g1 = {0x00020000u, 64u << 16, 0u, (64u << 16) | 8u,
                      8u, 64u, 0u, 0u};

#pragma unroll
    for (int d = 0; d < 4; ++d) {
        const int i = ibase + d;
        const uint64_t ga = (uint64_t)(uintptr_t)tables
                          + (uint64_t)i * ((uint64_t)BUCKETS * 256ull);
        // group0: count=1 | gather_index_size(32b)<<30 | gather_mode<<31,
        //         lds_addr, global_addr[31:0], addr[56:32] | type=2<<30
        const u32x4 g0 = {0xC0000001u,
                          sbufBase + (u32)(((tgrp << 3) | i) << 11),
                          (u32)ga,
                          ((u32)(ga >> 32) & 0x01FFFFFFu) | 0x80000000u};
        // groups 2+3: eight 32-bit row indices (lane (r<<2)|d holds row r).
        const u32x4 g2 = {(u32)__builtin_amdgcn_readlane((int)ia, (0 << 2) | d),
                          (u32)__builtin_amdgcn_readlane((int)ia, (1 << 2) | d),
                          (u32)__builtin_amdgcn_readlane((int)ia, (2 << 2) | d),
                          (u32)__builtin_amdgcn_readlane((int)ia, (3 << 2) | d)};
        const u32x4 g3 = {(u32)__builtin_amdgcn_readlane((int)ia, (4 << 2) | d),
                          (u32)__builtin_amdgcn_readlane((int)ia, (5 << 2) | d),
                          (u32)__builtin_amdgcn_readlane((int)ia, (6 << 2) | d),
                          (u32)__builtin_amdgcn_readlane((int)ia, (7 << 2) | d)};
        asm volatile("tensor_load_to_lds %0, %1, %2, %3"
                     :: "s"(g0), "s"(g1), "s"(g2), "s"(g3) : "memory");
    }
#if __has_builtin(__builtin_amdgcn_s_wait_tensorcnt)
    __builtin_amdgcn_s_wait_tensorcnt(0);
#else
    asm volatile("s_wait_tensorcnt 0x0" ::: "memory");
#endif
    __syncthreads();   // all 32 descriptors of this block are in LDS

    // ---- Copy-out: LDS -> out, async b128, nontemporal (keep L2 for tables).
    // Chunk q: c=q&15, r=(q>>4)&7, i=(q>>7)&7, tgrp=q>>10; LDS src = q*16.
    f4* out4 = (f4*)out;
    const size_t outBase4 = ((size_t)b * Tdim + t0) * (NUM_TABLES * EMBED / 4);
#pragma unroll 4
    for (int q = tid; q < NCHUNK; q += BLOCK) {
        const unsigned ldsOff = sbufBase + (unsigned)(q << 4);
        const int c  = q & 15;
        const int r  = (q >> 4) & 7;
        const int i  = (q >> 7) & 7;
        const int tg = q >> 10;
        f4* dst = out4 + outBase4
                + (size_t)((((tg << 3) + r) << 7) + (i << 4) + c);
        asm volatile("global_store_async_from_lds_b128 %0, %1, off th:TH_STORE_NT"
                     :: "v"(dst), "v"(ldsOff) : "memory");
    }
    asm volatile("s_wait_asynccnt 0x0" ::: "memory");
}

extern "C" void kernel_launch(void* const* d_in, const int* in_sizes, int n_in,
                              void* d_out, int out_size, void* d_ws, size_t ws_size,
                              hipStream_t stream) {
    (void)in_sizes; (void)n_in; (void)out_size; (void)d_ws; (void)ws_size;
    const int*   tokens = (const int*)d_in[0];
    const float* tables = (const float*)d_in[1];
    float*       out    = (float*)d_out;

    const int nBlocks = (Bdim * Tdim) / TILE_T;   // 1024
    HashTables_tdm_kernel<<<dim3(nBlocks), dim3(BLOCK), 0, stream>>>(
        tokens, tables, out);
}